// Rank_IGR_Loss_61091614819142
// MI455X (gfx1250) — compile-verified
//
#include <hip/hip_runtime.h>
#include <hip/hip_bf16.h>

#define MAXN    3136        // >= 3125, multiple of 32 and 16
#define THREADS 512
#define NWAVES  (THREADS / 32)

typedef float v2f __attribute__((ext_vector_type(2)));
typedef float v8f __attribute__((ext_vector_type(8)));

#if defined(__has_builtin)
#  if __has_builtin(__builtin_amdgcn_ballot_w32)
#    define BALLOT32(x) __builtin_amdgcn_ballot_w32(x)
#  endif
#endif
#ifndef BALLOT32
#  define BALLOT32(x) ((unsigned)__ballot(x))
#endif

__global__ __launch_bounds__(THREADS)
void rank_igr_loss_kernel(const float* __restrict__ cls,
                          const int*   __restrict__ lab,
                          const float* __restrict__ bbox,
                          const float* __restrict__ tgt,
                          float* __restrict__ ws, int B, int N) {
    __shared__ float sIou[MAXN];
    __shared__ float sP[MAXN];
    __shared__ int   sFlag[MAXN];
    __shared__ float sRed[THREADS];
    __shared__ int   sCount;

    const int b    = blockIdx.x;
    const int tid  = threadIdx.x;
    const int lane = tid & 31;
    const int wave = tid >> 5;
    const int hi   = lane >> 4;   // 0 or 1 (which M-half this lane's C/D rows map to)
    const int ln   = lane & 15;   // column N within tile

    const int NN = (N > MAXN) ? MAXN : N;

    const float tx1 = tgt[b * 4 + 0], ty1 = tgt[b * 4 + 1];
    const float tx2 = tgt[b * 4 + 2], ty2 = tgt[b * 4 + 3];
    const float tarea = (tx2 - tx1) * (ty2 - ty1);

    // ---- Phase 1: stage pos flag, p = exp(cls[:,1]), iou into LDS ----
    for (int i = tid; i < NN; i += THREADS) {
        int   l  = lab[b * N + i];
        float c1 = cls[(b * N + i) * 2 + 1];
        float x1 = bbox[(b * 4 + 0) * N + i];
        float y1 = bbox[(b * 4 + 1) * N + i];
        float x2 = bbox[(b * 4 + 2) * N + i];
        float y2 = bbox[(b * 4 + 3) * N + i];
        float wwi = fmaxf(fminf(tx2, x2) - fmaxf(tx1, x1), 0.0f);
        float hhi = fmaxf(fminf(ty2, y2) - fmaxf(ty1, y1), 0.0f);
        float area  = (x2 - x1) * (y2 - y1);
        float inter = wwi * hhi;
        sIou[i]  = inter / (area + tarea - inter);
        sP[i]    = expf(c1);
        sFlag[i] = (l > 0) ? 1 : 0;
    }
    __syncthreads();

    // ---- Phase 2: deterministic, order-preserving in-place compaction (wave 0) ----
    if (wave == 0) {
        int cnt = 0;
        for (int base = 0; base < NN; base += 32) {
            int  idx = base + lane;
            bool f   = (idx < NN) && (sFlag[idx] != 0);
            float vi = f ? sIou[idx] : 0.0f;
            float vp = f ? sP[idx]   : 0.0f;
            unsigned m = BALLOT32(f);
            int pre = __popc(m & ((1u << lane) - 1u));
            if (f) { sIou[cnt + pre] = vi; sP[cnt + pre] = vp; }  // dst <= src: safe in place
            cnt += __popc(m);
        }
        if (lane == 0) sCount = cnt;
    }
    __syncthreads();

    const int P = sCount;                 // number of positives (index order preserved)
    const int T = (P + 15) >> 4;          // 16-wide tiles

    // ---- Phase 3: WMMA-tiled pairwise sums ----
    // loss1 term for pair (gi<gj): iou_i >= iou_j ? a_i*b_j : a_j*b_i  (a=e^-p, b=e^p)
    // loss2 term for pair (gi<gj): p_i   >= p_j   ? c_i*d_j : c_j*d_i  (c=e^-iou, d=e^iou)
    float acc1 = 0.0f, acc2 = 0.0f;
    const v8f cz = {0, 0, 0, 0, 0, 0, 0, 0};

    for (int I = wave; I < T; I += NWAVES) {
        const int Ib = I << 4;
        // Cache this lane's 8 row keys (C/D layout: row m = v + 8*hi)
        float iouM[8], pM[8];
#pragma unroll
        for (int v = 0; v < 8; ++v) {
            int  gi = Ib + v + (hi << 3);
            bool ok = gi < P;
            iouM[v] = ok ? sIou[gi] : 0.0f;
            pM[v]   = ok ? sP[gi]   : 0.0f;
        }
        // A-matrix fragments: 16x4 f32, vector in column K=0 (lanes 0-15 hold K=0)
        int   gr  = Ib + ln;
        bool  rok = (gr < P) && (hi == 0);
        float pr  = rok ? sP[gr]   : 0.0f;
        float ir  = rok ? sIou[gr] : 0.0f;
        v2f Afa = { rok ? expf(-pr) : 0.0f, 0.0f };   // a_i
        v2f Afb = { rok ? expf( pr) : 0.0f, 0.0f };   // b_i
        v2f Afc = { rok ? expf(-ir) : 0.0f, 0.0f };   // c_i
        v2f Afd = { rok ? expf( ir) : 0.0f, 0.0f };   // d_i

        for (int J = I; J < T; ++J) {
            const int Jb = J << 4;
            int   gj  = Jb + ln;
            bool  jok = gj < P;
            float pj  = jok ? sP[gj]   : 0.0f;
            float ij  = jok ? sIou[gj] : 0.0f;
            bool  fok = jok && (hi == 0);
            // B-matrix fragments: 4x16 f32, vector in row K=0 (lanes 0-15 hold K=0)
            v2f Bfb = { fok ? expf( pj) : 0.0f, 0.0f };
            v2f Bfa = { fok ? expf(-pj) : 0.0f, 0.0f };
            v2f Bfd = { fok ? expf( ij) : 0.0f, 0.0f };
            v2f Bfc = { fok ? expf(-ij) : 0.0f, 0.0f };

            // Four 16x16 outer products on the matrix unit
            v8f Pm = __builtin_amdgcn_wmma_f32_16x16x4_f32(false, Afa, false, Bfb, (short)0, cz, false, false);
            v8f Qm = __builtin_amdgcn_wmma_f32_16x16x4_f32(false, Afb, false, Bfa, (short)0, cz, false, false);
            v8f Rm = __builtin_amdgcn_wmma_f32_16x16x4_f32(false, Afc, false, Bfd, (short)0, cz, false, false);
            v8f Sm = __builtin_amdgcn_wmma_f32_16x16x4_f32(false, Afd, false, Bfc, (short)0, cz, false, false);

#pragma unroll
            for (int v = 0; v < 8; ++v) {
                int  gi = Ib + v + (hi << 3);
                bool g  = gi < gj;                      // count each unordered pair once
                float t1 = (iouM[v] >= ij) ? Pm[v] : Qm[v];
                float t2 = (pM[v]   >= pj) ? Rm[v] : Sm[v];
                acc1 += g ? t1 : 0.0f;
                acc2 += g ? t2 : 0.0f;
            }
        }
    }

    // ---- Phase 4: deterministic block tree reduction ----
    sRed[tid] = acc1;
    __syncthreads();
    for (int s = THREADS / 2; s > 0; s >>= 1) {
        if (tid < s) sRed[tid] += sRed[tid + s];
        __syncthreads();
    }
    float l1 = sRed[0];
    __syncthreads();
    sRed[tid] = acc2;
    __syncthreads();
    for (int s = THREADS / 2; s > 0; s >>= 1) {
        if (tid < s) sRed[tid] += sRed[tid + s];
        __syncthreads();
    }
    if (tid == 0) {
        float l2 = sRed[0];
        long long np = (long long)P * (long long)(P - 1) / 2;
        float denom = (float)((np > 0) ? np : 1);
        ws[b * 3 + 0] = l1 / denom;
        ws[b * 3 + 1] = l2 / denom;
        ws[b * 3 + 2] = (np > 0) ? 1.0f : 0.0f;
    }
}

__global__ void rank_igr_finalize_kernel(const float* __restrict__ ws,
                                         float* __restrict__ out, int B) {
    if (blockIdx.x == 0 && threadIdx.x == 0) {
        float s1 = 0.0f, s2 = 0.0f, nv = 0.0f;
        for (int b = 0; b < B; ++b) {
            float v = ws[b * 3 + 2];
            s1 += ws[b * 3 + 0] * v;
            s2 += ws[b * 3 + 1] * v;
            nv += v;
        }
        float d = fmaxf(nv, 1.0f);
        out[0] = s1 / d;
        out[1] = s2 / d;
    }
}

extern "C" void kernel_launch(void* const* d_in, const int* in_sizes, int n_in,
                              void* d_out, int out_size, void* d_ws, size_t ws_size,
                              hipStream_t stream) {
    const float* cls  = (const float*)d_in[0];
    const int*   lab  = (const int*)d_in[1];
    const float* bbox = (const float*)d_in[2];
    const float* tgt  = (const float*)d_in[3];
    const int B = in_sizes[3] / 4;          // label_target is [B,4]
    const int N = in_sizes[1] / B;          // label_cls is [B,N]
    float* ws = (float*)d_ws;

    rank_igr_loss_kernel<<<B, THREADS, 0, stream>>>(cls, lab, bbox, tgt, ws, B, N);
    rank_igr_finalize_kernel<<<1, 32, 0, stream>>>(ws, (float*)d_out, B);
}